// DGMN2Attention_50130858279073
// MI455X (gfx1250) — compile-verified
//
#include <hip/hip_runtime.h>
#include <hip/hip_bf16.h>

typedef __attribute__((ext_vector_type(2))) float v2f;
typedef __attribute__((ext_vector_type(8))) float v8f;

#define HH    56
#define WW    56
#define NPIX  (HH * WW)     // 3136
#define CC    256
#define NHEAD 8
#define DHEAD 32
#define NSAMP 9

// ---------------------------------------------------------------------------
// WMMA fp32 GEMM:  C(MxN) = A(MxK) * B(KxN) (+ optional bias per column)
// Block = 256 threads = 8 wave32s arranged 4(M) x 2(N).
// Block tile: 64M x 128N. Each wave: one 16-row stripe x 64 cols =
// four 16x16 accumulators, reusing a single A fragment per K-step
// (4x V_WMMA_F32_16X16X4_F32 per 4-wide K-step -> ~6.4 FLOP/B from cache).
// Fragment layout (ISA 7.12.2):
//   A 16x4 : lane = m + 16*(k>=2), vgpr = k&1  -> contiguous float2 per lane
//   B 4x16 : lane = n + 16*(k>=2), vgpr = k&1
//   C 16x16: lane = n + 16*(m>=8), vgpr = m&7
// ---------------------------------------------------------------------------
__global__ __launch_bounds__(256) void wmma_gemm_f32(
    const float* __restrict__ A, const float* __restrict__ B,
    const float* __restrict__ bias, float* __restrict__ C,
    int M, int N, int K)
{
  const int wave = threadIdx.x >> 5;      // 0..7
  const int lane = threadIdx.x & 31;
  const int half = lane >> 4;             // K-pair / M-half selector
  const int mn   = lane & 15;             // A: m, B/C: n
  const int wm   = wave & 3;              // wave's M slot (0..3)
  const int wn   = wave >> 2;             // wave's N slot (0..1)
  const int row0 = (blockIdx.y << 6) + (wm << 4);
  const int col0 = (blockIdx.x << 7) + (wn << 6);  // 64-col wave stripe
  if (row0 >= M || col0 >= N) return;     // wave-uniform -> EXEC stays all-1s

  v8f acc0 = {}, acc1 = {}, acc2 = {}, acc3 = {};
  const float* Arow  = A + (size_t)(row0 + mn) * K;
  const float* Bbase = B + col0 + mn;

  #pragma unroll 2
  for (int k0 = 0; k0 < K; k0 += 4) {
    const int ka = k0 + (half << 1);
    // prefetch next A K-panel (global_prefetch_b8), hidden under WMMA pipe
    __builtin_prefetch(Arow + ka + 32, 0, 1);

    v2f a;
    a.x = Arow[ka];
    a.y = Arow[ka + 1];

    const float* B0 = Bbase + (size_t)ka * N;        // row ka
    const float* B1 = Bbase + (size_t)(ka + 1) * N;  // row ka+1
    v2f b0, b1, b2, b3;
    b0.x = B0[0];  b0.y = B1[0];
    b1.x = B0[16]; b1.y = B1[16];
    b2.x = B0[32]; b2.y = B1[32];
    b3.x = B0[48]; b3.y = B1[48];

    acc0 = __builtin_amdgcn_wmma_f32_16x16x4_f32(false, a, false, b0,
                                                 (short)0, acc0, false, false);
    acc1 = __builtin_amdgcn_wmma_f32_16x16x4_f32(false, a, false, b1,
                                                 (short)0, acc1, false, false);
    acc2 = __builtin_amdgcn_wmma_f32_16x16x4_f32(false, a, false, b2,
                                                 (short)0, acc2, false, false);
    acc3 = __builtin_amdgcn_wmma_f32_16x16x4_f32(false, a, false, b3,
                                                 (short)0, acc3, false, false);
  }

  #pragma unroll
  for (int t = 0; t < 4; ++t) {
    const v8f acc = (t == 0) ? acc0 : (t == 1) ? acc1 : (t == 2) ? acc2 : acc3;
    const int col = col0 + (t << 4) + mn;
    const float bv = bias ? bias[col] : 0.0f;
    #pragma unroll
    for (int i = 0; i < 8; ++i) {
      const int m = (half << 3) + i;
      C[(size_t)(row0 + m) * N + col] = acc[i] + bv;
    }
  }
}

// ---------------------------------------------------------------------------
// Per-(b,head,pixel) offset projection: off[t, 0..17] = x_head(32) @ off_w(32x18)
// ---------------------------------------------------------------------------
__global__ __launch_bounds__(256) void off_kernel(
    const float* __restrict__ x, const float* __restrict__ off_w,
    float* __restrict__ off, int B)
{
  const int t = blockIdx.x * blockDim.x + threadIdx.x;  // t = bh*NPIX + n
  if (t >= B * NHEAD * NPIX) return;
  const int n  = t % NPIX;
  const int bh = t / NPIX;
  const int b = bh >> 3, head = bh & 7;
  const float* xr = x + ((size_t)(b * NPIX + n)) * CC + head * DHEAD;

  float o[18];
  #pragma unroll
  for (int e = 0; e < 18; ++e) o[e] = 0.0f;
  #pragma unroll 8
  for (int dd = 0; dd < DHEAD; ++dd) {
    const float xv = xr[dd];
    const float* wr = off_w + dd * 18;
    #pragma unroll
    for (int e = 0; e < 18; ++e) o[e] = fmaf(xv, wr[e], o[e]);
  }
  float* orow = off + (size_t)t * 18;
  #pragma unroll
  for (int e = 0; e < 18; ++e) orow[e] = o[e];
}

// ---------------------------------------------------------------------------
// Fused deformable sampling + rel-pos logits + softmax + weighted-V.
// One wave32 per (b, head, pixel); lane = channel dd (d == 32 == wave size).
// k/v gathers are 32 consecutive floats in qkv -> coalesced 128B per corner.
// ---------------------------------------------------------------------------
__global__ __launch_bounds__(256) void sample_attend_kernel(
    const float* __restrict__ qkv,   // (B*NPIX, 768)  q | k | v
    const float* __restrict__ off,   // (B*h*NPIX, 18) interleaved dy,dx per sample
    const float* __restrict__ relh,  // (64, 32)
    const float* __restrict__ relw,  // (64, 32)
    float* __restrict__ outp,        // (B*NPIX, 256)
    int B)
{
  const int wave = threadIdx.x >> 5;
  const int lane = threadIdx.x & 31;              // channel dd
  const int bh = blockIdx.y;
  const int n  = blockIdx.x * 8 + wave;
  const int b = bh >> 3, head = bh & 7;
  const int xp = n / WW, yp = n % WW;             // pixel (H-idx, W-idx)
  const int nt = yp * WW + xp;                    // transposed pixel

  const float q_dd  = qkv[((size_t)(b * NPIX + n )) * 768 + head * DHEAD + lane];
  const float q2_dd = qkv[((size_t)(b * NPIX + nt)) * 768 + head * DHEAD + lane];
  const float scale = 0.17677669529663687f;       // 32^-0.5

  const float* offr = off + ((size_t)bh * NPIX + n) * 18;

  float vs[NSAMP];
  float logit[NSAMP];

  #pragma unroll
  for (int j = 0; j < NSAMP; ++j) {
    const float dy = offr[2 * j];
    const float dx = offr[2 * j + 1];
    const float ysf = dy + (float)xp + (float)(j / 3 - 1);
    const float xsf = dx + (float)yp + (float)(j % 3 - 1);
    const float y0 = floorf(ysf), x0 = floorf(xsf);
    const float ly = ysf - y0,    lx = xsf - x0;
    const float wts[4] = { (1.f - ly) * (1.f - lx), (1.f - ly) * lx,
                           ly * (1.f - lx),         ly * lx };
    float ks_dd = 0.f, vs_dd = 0.f;
    #pragma unroll
    for (int c = 0; c < 4; ++c) {
      const float yc = y0 + (float)(c >> 1);
      const float xc = x0 + (float)(c & 1);
      const float valid = (yc >= 0.f && yc <= (float)(HH - 1) &&
                           xc >= 0.f && xc <= (float)(WW - 1)) ? 1.f : 0.f;
      const int yi = (int)fminf(fmaxf(yc, 0.f), (float)(HH - 1));
      const int xi = (int)fminf(fmaxf(xc, 0.f), (float)(WW - 1));
      const size_t pbase = ((size_t)(b * NPIX + yi * WW + xi)) * 768
                         + head * DHEAD + lane;
      ks_dd = fmaf(qkv[pbase + 256] * valid, wts[c], ks_dd);  // k gather
      vs_dd = fmaf(qkv[pbase + 512] * valid, wts[c], vs_dd);  // v gather
    }
    vs[j] = vs_dd;
    // rel_to_abs closed form; reference adds h_logits WITHOUT re-transpose:
    //   pos[j] = q(xp,yp).relw[55+j-yp] + q(yp,xp).relh[55+j-yp]
    float t = q_dd * ks_dd * scale
            + q_dd  * relw[(55 + j - yp) * DHEAD + lane]
            + q2_dd * relh[(55 + j - yp) * DHEAD + lane];
    #pragma unroll
    for (int m = 16; m > 0; m >>= 1) t += __shfl_xor(t, m, 32);
    logit[j] = t;                                  // full sum in every lane
  }

  // softmax over 9 samples (identical in all lanes)
  float mx = logit[0];
  #pragma unroll
  for (int j = 1; j < NSAMP; ++j) mx = fmaxf(mx, logit[j]);
  float sum = 0.f;
  #pragma unroll
  for (int j = 0; j < NSAMP; ++j) { logit[j] = __expf(logit[j] - mx); sum += logit[j]; }
  const float inv = 1.f / sum;

  float o = 0.f;
  #pragma unroll
  for (int j = 0; j < NSAMP; ++j) o = fmaf(logit[j] * inv, vs[j], o);

  outp[((size_t)(b * NPIX + n)) * CC + head * DHEAD + lane] = o;
}

// ---------------------------------------------------------------------------
extern "C" void kernel_launch(void* const* d_in, const int* in_sizes, int n_in,
                              void* d_out, int out_size, void* d_ws, size_t ws_size,
                              hipStream_t stream)
{
  const float* x      = (const float*)d_in[0];
  const float* qkv_w  = (const float*)d_in[1];
  const float* off_w  = (const float*)d_in[2];
  const float* proj_w = (const float*)d_in[3];
  const float* proj_b = (const float*)d_in[4];
  const float* relh   = (const float*)d_in[5];
  const float* relw   = (const float*)d_in[6];

  const int B = in_sizes[0] / (NPIX * CC);   // 4
  const int M = B * NPIX;                    // 12544 (multiple of 64)

  // workspace carve-up (fp32): qkv (M x 768) | off (B*h*N x 18) | out_pre (M x 256)
  float* qkv  = (float*)d_ws;
  float* off  = qkv + (size_t)M * 768;
  float* outp = off + (size_t)B * NHEAD * NPIX * 18;

  // 1) qkv = x @ qkv_w                (WMMA f32, 4.9 GFLOP)
  dim3 g1(3 * CC / 128, M / 64);             // 6 x 196 blocks
  wmma_gemm_f32<<<g1, 256, 0, stream>>>(x, qkv_w, nullptr, qkv, M, 3 * CC, CC);

  // 2) deform offsets
  const int total = B * NHEAD * NPIX;
  off_kernel<<<(total + 255) / 256, 256, 0, stream>>>(x, off_w, off, B);

  // 3) fused sampling + rel-pos + softmax + weighted V
  dim3 g3(NPIX / 8, B * NHEAD);
  sample_attend_kernel<<<g3, 256, 0, stream>>>(qkv, off, relh, relw, outp, B);

  // 4) out = out_pre @ proj_w + proj_b   (WMMA f32, 1.6 GFLOP)
  dim3 g4(CC / 128, M / 64);                 // 2 x 196 blocks
  wmma_gemm_f32<<<g4, 256, 0, stream>>>(outp, proj_w, proj_b, (float*)d_out,
                                        M, CC, CC);
}